// Processor_481036337792
// MI455X (gfx1250) — compile-verified
//
#include <hip/hip_runtime.h>

// GNN Processor: 4x GraphConv (gather-scale -> scatter-add -> dual GEMM + bias
// + ReLU (+ residual)) on MI455X (gfx1250, wave32).
//
// Phase kernels:
//   zero_kernel     : clear the aggregation buffer
//   scatter_kernel  : agg[dst] += w_e * x[src]   (float4 gather, native fp32
//                     L2 atomics -- whole working set fits in 192MB L2)
//   gemm_wmma_kernel: out = relu(agg@Wrel^T + b + x@Wroot^T) [+relu residual]
//                     using V_WMMA_F32_16X16X4_F32 (fp32-exact matrix path)

#define DH 128

typedef __attribute__((ext_vector_type(2))) float v2f;
typedef __attribute__((ext_vector_type(8))) float v8f;

__global__ __launch_bounds__(256)
void zero_kernel(float* __restrict__ p, int n) {
    int i = blockIdx.x * blockDim.x + threadIdx.x;
    if (i < n) p[i] = 0.0f;
}

// One thread per (edge, 4-feature chunk).
template <int D>
__global__ __launch_bounds__(256)
void scatter_kernel(const float* __restrict__ x,
                    const int*   __restrict__ ei,   // [2,E]: src row then dst row
                    const float* __restrict__ w,
                    float*       __restrict__ agg,
                    int E) {
    constexpr int CH = D / 4;
    int tid = blockIdx.x * blockDim.x + threadIdx.x;
    if (tid >= E * CH) return;
    int e = tid / CH;
    int c = (tid - e * CH) * 4;
    int s = ei[e];
    int d = ei[E + e];
    float we = w[e];
    const float4 v = *(const float4*)(x + (size_t)s * D + c);
    float* dp = agg + (size_t)d * D + c;
    unsafeAtomicAdd(dp + 0, v.x * we);
    unsafeAtomicAdd(dp + 1, v.y * we);
    unsafeAtomicAdd(dp + 2, v.z * we);
    unsafeAtomicAdd(dp + 3, v.w * we);
}

// One wave32 per 16x16 output tile.
// out[m][n] = relu( sum_k agg[m][k]*Wrel[n][k] + brel[n] + sum_k x[m][k]*Wroot[n][k] )
// if SKIP:  out = relu(out + x[m][n])   (residual; only valid when DIN==DH)
//
// f32 WMMA operand layout (ISA 7.12.2):
//   A (16x4): lane = (Khalf<<4)|M,  vgpr j -> K = 2*Khalf + j
//   B (4x16): lane = (Khalf<<4)|N,  vgpr j -> K = 2*Khalf + j
//   C/D     : lane = (Mhalf<<4)|N,  vgpr r -> M = 8*Mhalf + r
// Both A (row-major MxK) and W (row-major NxK) therefore load as one float2
// per lane per K-step with identical index math.
template <int DIN, bool SKIP>
__global__ __launch_bounds__(256)
void gemm_wmma_kernel(const float* __restrict__ agg,
                      const float* __restrict__ x,
                      const float* __restrict__ Wrel,
                      const float* __restrict__ brel,
                      const float* __restrict__ Wroot,
                      float*       __restrict__ out,
                      int nNodes) {
    constexpr int NT = DH / 16;                       // 8 column tiles
    int wave    = (blockIdx.x * blockDim.x + threadIdx.x) >> 5;
    int mtile   = wave / NT;
    int ntile   = wave - mtile * NT;
    int rowBase = mtile * 16;
    if (rowBase >= nNodes) return;                    // wave-uniform
    int colBase = ntile * 16;

    int lane = threadIdx.x & 31;
    int ln   = lane & 15;                             // M (for A) / N (for B)
    int kh   = (lane >> 4) * 2;                       // K sub-offset

    const float* ap = agg   + (size_t)(rowBase + ln) * DIN + kh;
    const float* xp = x     + (size_t)(rowBase + ln) * DIN + kh;
    const float* rp = Wrel  + (size_t)(colBase + ln) * DIN + kh;
    const float* op = Wroot + (size_t)(colBase + ln) * DIN + kh;

    v8f acc = {};
#pragma unroll
    for (int k0 = 0; k0 < DIN; k0 += 4) {
        v2f a  = *(const v2f*)(ap + k0);
        v2f b  = *(const v2f*)(rp + k0);
        acc = __builtin_amdgcn_wmma_f32_16x16x4_f32(false, a, false, b,
                                                    (short)0, acc, false, false);
        v2f a2 = *(const v2f*)(xp + k0);
        v2f b2 = *(const v2f*)(op + k0);
        acc = __builtin_amdgcn_wmma_f32_16x16x4_f32(false, a2, false, b2,
                                                    (short)0, acc, false, false);
    }

    float bias = brel[colBase + ln];
    int rbase  = rowBase + (lane >> 4) * 8;
#pragma unroll
    for (int r = 0; r < 8; ++r) {
        int row = rbase + r;
        float h = fmaxf(acc[r] + bias, 0.0f);
        if (SKIP)
            h = fmaxf(h + x[(size_t)row * DH + colBase + ln], 0.0f);
        out[(size_t)row * DH + colBase + ln] = h;
    }
}

extern "C" void kernel_launch(void* const* d_in, const int* in_sizes, int n_in,
                              void* d_out, int out_size, void* d_ws, size_t ws_size,
                              hipStream_t stream) {
    const float* z      = (const float*)d_in[0];
    const int*   ei     = (const int*)  d_in[1];   // [2,E]
    const float* ew     = (const float*)d_in[2];
    /* d_in[3] = batch, unused */
    const float* Wrel0  = (const float*)d_in[4];   // [128,32]
    const float* brel0  = (const float*)d_in[5];   // [128]
    const float* Wroot0 = (const float*)d_in[6];   // [128,32]
    const float* Wrel   = (const float*)d_in[7];   // [3,128,128]
    const float* brel   = (const float*)d_in[8];   // [3,128]
    const float* Wroot  = (const float*)d_in[9];   // [3,128,128]

    const int N = in_sizes[0] / 32;
    const int E = in_sizes[2];

    float* outp = (float*)d_out;
    float* bufA = (float*)d_ws;                  // [N,128]
    float* bufB = bufA + (size_t)N * DH;         // [N,128]
    float* agg  = bufB + (size_t)N * DH;         // [N,128] aggregation scratch

    const int mtiles = (N + 15) / 16;
    const int gemmGrid = (mtiles * (DH / 16) + 7) / 8;   // 8 waves/block

    auto zero = [&](float* p, int n) {
        zero_kernel<<<(n + 255) / 256, 256, 0, stream>>>(p, n);
    };

    // ---- GBlock 0: 32 -> 128, no skip: z -> bufA
    zero(agg, N * 32);
    scatter_kernel<32><<<(E * 8 + 255) / 256, 256, 0, stream>>>(z, ei, ew, agg, E);
    gemm_wmma_kernel<32, false><<<gemmGrid, 256, 0, stream>>>(
        agg, z, Wrel0, brel0, Wroot0, bufA, N);

    // ---- GBlock 1: 128 -> 128, skip: bufA -> bufB
    zero(agg, N * DH);
    scatter_kernel<128><<<(E * 32 + 255) / 256, 256, 0, stream>>>(bufA, ei, ew, agg, E);
    gemm_wmma_kernel<128, true><<<gemmGrid, 256, 0, stream>>>(
        agg, bufA, Wrel + 0 * DH * DH, brel + 0 * DH, Wroot + 0 * DH * DH, bufB, N);

    // ---- GBlock 2: 128 -> 128, skip: bufB -> bufA
    zero(agg, N * DH);
    scatter_kernel<128><<<(E * 32 + 255) / 256, 256, 0, stream>>>(bufB, ei, ew, agg, E);
    gemm_wmma_kernel<128, true><<<gemmGrid, 256, 0, stream>>>(
        agg, bufB, Wrel + 1 * DH * DH, brel + 1 * DH, Wroot + 1 * DH * DH, bufA, N);

    // ---- GBlock 3: 128 -> 128, no skip: bufA -> d_out
    zero(agg, N * DH);
    scatter_kernel<128><<<(E * 32 + 255) / 256, 256, 0, stream>>>(bufA, ei, ew, agg, E);
    gemm_wmma_kernel<128, false><<<gemmGrid, 256, 0, stream>>>(
        agg, bufA, Wrel + 2 * DH * DH, brel + 2 * DH, Wroot + 2 * DH * DH, outp, N);
}